// Prompt_70068096467863
// MI455X (gfx1250) — compile-verified
//
#include <hip/hip_runtime.h>

// Problem constants from the reference
static constexpr int kB    = 128;   // batch
static constexpr int kS    = 197;   // sequence
static constexpr int kD    = 768;   // embed dim
static constexpr int kL    = 5;     // layers
static constexpr int kP    = 20;    // pool size
static constexpr int kPpad = 32;    // pool size padded to 2 WMMA N-tiles
static constexpr int kPLEN = 5;     // prompt length
static constexpr int kK    = 5;     // top-k

typedef float v2f __attribute__((ext_vector_type(2)));
typedef float v8f __attribute__((ext_vector_type(8)));

// ---------------------------------------------------------------------------
// Kernel 1: x_norm[b, :] = l2_normalize(mean_s x[b, s, :])
// One block per batch row; 192 threads, one float4 column-group each.
// The 77 MB x_embed read becomes pure global_load_b128 streams.
// ---------------------------------------------------------------------------
__global__ void mean_norm_x_kernel(const float* __restrict__ x,
                                   float* __restrict__ xnorm) {
    __shared__ float red[192];
    __shared__ float s_inv;
    const int b = blockIdx.x;
    const int t = threadIdx.x;          // 0..191, column group = 4*t

    const float4* base = (const float4*)(x + (size_t)b * kS * kD);
    float4 acc = make_float4(0.f, 0.f, 0.f, 0.f);
    for (int s = 0; s < kS; ++s) {
        float4 v = base[(size_t)s * (kD / 4) + t];   // b128 load
        acc.x += v.x; acc.y += v.y; acc.z += v.z; acc.w += v.w;
    }
    const float inv_s = 1.0f / (float)kS;
    acc.x *= inv_s; acc.y *= inv_s; acc.z *= inv_s; acc.w *= inv_s;

    red[t] = acc.x * acc.x + acc.y * acc.y + acc.z * acc.z + acc.w * acc.w;
    __syncthreads();
    for (int off = 96; off > 0; off >>= 1) {
        if (t < off) red[t] += red[t + off];
        __syncthreads();
    }
    if (t == 0) s_inv = rsqrtf(fmaxf(red[0], 1e-12f));
    __syncthreads();

    const float inv = s_inv;
    float4* o = (float4*)(xnorm + (size_t)b * kD);
    o[t] = make_float4(acc.x * inv, acc.y * inv, acc.z * inv, acc.w * inv);
}

// ---------------------------------------------------------------------------
// Kernel 2: pk_norm[32][768]: rows 0..19 = l2_normalize(prompt_key),
// rows 20..31 = zeros. Zero-padding here makes the WMMA inner loop branch-free
// (no EXEC-predicated loads, which dominated round-1 codegen).
// ---------------------------------------------------------------------------
__global__ void norm_pk_padded_kernel(const float* __restrict__ pk,
                                      float* __restrict__ pknorm) {
    __shared__ float red[192];
    __shared__ float s_inv;
    const int p = blockIdx.x;           // 0..31
    const int t = threadIdx.x;          // 0..191

    float4* o = (float4*)(pknorm + (size_t)p * kD);
    if (p >= kP) {
        o[t] = make_float4(0.f, 0.f, 0.f, 0.f);
        return;
    }

    const float4 v = ((const float4*)(pk + (size_t)p * kD))[t];
    red[t] = v.x * v.x + v.y * v.y + v.z * v.z + v.w * v.w;
    __syncthreads();
    for (int off = 96; off > 0; off >>= 1) {
        if (t < off) red[t] += red[t + off];
        __syncthreads();
    }
    if (t == 0) s_inv = rsqrtf(fmaxf(red[0], 1e-12f));
    __syncthreads();

    const float inv = s_inv;
    o[t] = make_float4(v.x * inv, v.y * inv, v.z * inv, v.w * inv);
}

// ---------------------------------------------------------------------------
// Kernel 3: sim[128,20] = x_norm @ pk_norm^T via V_WMMA_F32_16X16X4_F32.
// 16 waves: 8 M-tiles x 2 N-tiles over the zero-padded 32-row B matrix.
// K-loop: 192 WMMAs per wave, fp32 throughout (top-k ranking must be exact).
// Branch-free: unconditional global_load_b64 for A and B fragments.
// Fragment layouts per ISA 7.12.2:
//   A 16x4 f32: lanes 0-15 -> M rows, VGPR{0,1} = K{0,1}; lanes 16-31 = K{2,3}.
//   B 4x16 f32: row striped across lanes; lane halves mirror A's K split.
//   C/D 16x16:  VGPR v -> M = v (lanes 0-15) / v+8 (lanes 16-31), N = lane%16.
// ---------------------------------------------------------------------------
__global__ void sim_wmma_kernel(const float* __restrict__ xnorm,
                                const float* __restrict__ pknorm,
                                float* __restrict__ sim_ws,
                                float* __restrict__ sim_out) {
    const int tile = blockIdx.x;      // 0..15
    const int m0   = (tile >> 1) * 16;
    const int n0   = (tile & 1) * 16;
    const int lane = threadIdx.x;     // wave32
    const int hf   = lane >> 4;       // K half (0 or 1)
    const int lsub = lane & 15;

    const int n = n0 + lsub;                                   // 0..31, padded
    const float* arow = xnorm  + (size_t)(m0 + lsub) * kD + 2 * hf;
    const float* brow = pknorm + (size_t)n * kD + 2 * hf;

    v8f acc = {0.f, 0.f, 0.f, 0.f, 0.f, 0.f, 0.f, 0.f};
#pragma unroll 4
    for (int k = 0; k < kD; k += 4) {
        v2f a, bv;
        a.x  = arow[k];
        a.y  = arow[k + 1];
        bv.x = brow[k];
        bv.y = brow[k + 1];
        acc = __builtin_amdgcn_wmma_f32_16x16x4_f32(
                /*neg_a=*/false, a, /*neg_b=*/false, bv,
                /*c_mod=*/(short)0, acc, /*reuse_a=*/false, /*reuse_b=*/false);
    }

    if (n < kP) {
#pragma unroll
        for (int v = 0; v < 8; ++v) {
            const int m = m0 + v + 8 * hf;
            const float val = acc[v];
            sim_ws[m * kP + n]  = val;
            sim_out[m * kP + n] = val;
        }
    }
}

// ---------------------------------------------------------------------------
// Kernel 4: per-row top-5 of sim (strict > matches lax.top_k tie-breaking),
// plus reduce_sim = sum(top-k sims) / B (identical to sum(pk_norm[idx]*x_norm)).
// One block of 128 threads; one thread per batch row; LDS reduction for rsim.
// ---------------------------------------------------------------------------
__global__ void topk_kernel(const float* __restrict__ sim_ws,
                            int* __restrict__ idx_ws,
                            float* __restrict__ idx_out,
                            float* __restrict__ rsim_out) {
    __shared__ float red[128];
    const int b = threadIdx.x;

    float s[kP];
    const float* row = sim_ws + b * kP;
#pragma unroll
    for (int p = 0; p < kP; ++p) s[p] = row[p];

    const float NEG = -3.402823466e+38f;
    float sumtop = 0.f;
#pragma unroll
    for (int k = 0; k < kK; ++k) {
        float best = NEG;
        int bi = 0;
#pragma unroll
        for (int p = 0; p < kP; ++p) {
            if (s[p] > best) { best = s[p]; bi = p; }
        }
        s[bi] = NEG;
        idx_ws[b * kK + k]  = bi;
        idx_out[b * kK + k] = (float)bi;
        sumtop += best;
    }

    red[b] = sumtop;
    __syncthreads();
    for (int off = 64; off > 0; off >>= 1) {
        if (b < off) red[b] += red[b + off];
        __syncthreads();
    }
    if (b == 0) rsim_out[0] = red[0] / (float)kB;
}

// ---------------------------------------------------------------------------
// Kernel 5: batched_prompt gather, the 98.3 MB store stream.
// Output layout (L, 2, B, K*PLEN, H*DH=768); inner 768 floats contiguous.
// float4 (global b128) moves; source prompt (3.1 MB) stays L2-resident.
// Total = 6,144,000 float4s = 24000 blocks x 256 threads exactly.
// ---------------------------------------------------------------------------
__global__ void gather_prompt_kernel(const float4* __restrict__ prompt4,
                                     const int* __restrict__ idx_ws,
                                     float4* __restrict__ out4) {
    const int D4 = kD / 4;  // 192
    const long total = (long)kL * 2 * kB * (kK * kPLEN) * D4;
    long tid = (long)blockIdx.x * blockDim.x + threadIdx.x;
    if (tid >= total) return;

    const int c4 = (int)(tid % D4);
    long r = tid / D4;
    const int j = (int)(r % (kK * kPLEN)); r /= (kK * kPLEN);
    const int b = (int)(r % kB);           r /= kB;
    const int dual = (int)(r % 2);
    const int l    = (int)(r / 2);

    const int p  = idx_ws[b * kK + j / kPLEN];
    const int pl = j % kPLEN;

    const long src = ((((long)(l * 2 + dual) * kP + p) * kPLEN + pl) * D4) + c4;
    out4[tid] = prompt4[src];
}

// ---------------------------------------------------------------------------
// Launch. d_out layout (reference return order, flat):
//   [0, 24576000)              batched_prompt (L,2,B,25,12,64)
//   [24576000]                 reduce_sim
//   [24576001, 24578561)       sim (B,P)
//   [24578561, 24579201)       idx (B,K) as float
// ---------------------------------------------------------------------------
extern "C" void kernel_launch(void* const* d_in, const int* in_sizes, int n_in,
                              void* d_out, int out_size, void* d_ws, size_t ws_size,
                              hipStream_t stream) {
    const float* x      = (const float*)d_in[0];   // [128,197,768]
    const float* prompt = (const float*)d_in[1];   // [5,2,20,5,12,64]
    const float* pkey   = (const float*)d_in[2];   // [20,768]

    float* out = (float*)d_out;

    // Workspace carve-up (~505 KB of floats)
    float* xnorm  = (float*)d_ws;                     // 128*768
    float* pknorm = xnorm + (size_t)kB * kD;          // 32*768 (zero-padded)
    float* sim_ws = pknorm + (size_t)kPpad * kD;      // 128*20
    int*   idx_ws = (int*)(sim_ws + (size_t)kB * kP); // 128*5 ints

    const size_t NPROMPT = (size_t)kL * 2 * kB * (kK * kPLEN) * kD; // 24,576,000
    float* out_prompt = out;
    float* out_rsim   = out + NPROMPT;
    float* out_sim    = out_rsim + 1;
    float* out_idx    = out_sim + (size_t)kB * kP;

    mean_norm_x_kernel<<<kB, 192, 0, stream>>>(x, xnorm);
    norm_pk_padded_kernel<<<kPpad, 192, 0, stream>>>(pkey, pknorm);
    sim_wmma_kernel<<<16, 32, 0, stream>>>(xnorm, pknorm, sim_ws, out_sim);
    topk_kernel<<<1, 128, 0, stream>>>(sim_ws, idx_ws, out_idx, out_rsim);

    const long n4 = (long)(NPROMPT / 4);          // 6,144,000
    const int blocks = (int)((n4 + 255) / 256);   // 24000
    gather_prompt_kernel<<<blocks, 256, 0, stream>>>(
        (const float4*)prompt, idx_ws, (float4*)out_prompt);
}